// RecyclingEmbedder_38972533244445
// MI455X (gfx1250) — compile-verified
//
#include <hip/hip_runtime.h>

typedef float v2f __attribute__((ext_vector_type(2)));
typedef float v4f __attribute__((ext_vector_type(4)));
typedef float v8f __attribute__((ext_vector_type(8)));

#define C_M   256
#define C_Z   128
#define NRES  768
#define SSEQ  128
#define NBIN  15
#define LN_EPS 1e-5f
#define BIN_START 3.25f
#define BIN_STEP  1.25f

// full-wave (wave32) sum reduction
__device__ __forceinline__ float wave_sum32(float v) {
    v += __shfl_xor(v, 1, 32);
    v += __shfl_xor(v, 2, 32);
    v += __shfl_xor(v, 4, 32);
    v += __shfl_xor(v, 8, 32);
    v += __shfl_xor(v, 16, 32);
    return v;
}

// ---------------------------------------------------------------------------
// Kernel 1: stream-copy m[1:, :, :] -> out (float4, non-temporal).
// grid*block*4 floats == (SSEQ-1)*NRES*C_M exactly.
// ---------------------------------------------------------------------------
__global__ void __launch_bounds__(256)
m_copy_kernel(const float* __restrict__ m, float* __restrict__ out) {
    size_t idx = (size_t)blockIdx.x * blockDim.x + threadIdx.x;       // float4 idx
    size_t off = (size_t)(NRES * C_M / 4) + idx;                      // skip s==0
    v4f v = __builtin_nontemporal_load((const v4f*)m + off);
    __builtin_nontemporal_store(v, (v4f*)out + off);
}

// ---------------------------------------------------------------------------
// Kernel 2: out[0,n,:] = m[0,n,:] + LayerNorm(m_prev[0,n,:]) * w + b
// One wave per row n (C_M=256 -> 8 floats/lane). 96 blocks x 256 thr = 768 waves.
// ---------------------------------------------------------------------------
__global__ void __launch_bounds__(256)
m0_ln_kernel(const float* __restrict__ m, const float* __restrict__ m_prev,
             const float* __restrict__ w, const float* __restrict__ b,
             float* __restrict__ out) {
    const int lane = threadIdx.x & 31;
    const int n    = (blockIdx.x * blockDim.x + threadIdx.x) >> 5;    // row id
    const int c0   = lane * 8;
    const float* mp = m_prev + (size_t)n * C_M;

    v4f a0 = *(const v4f*)(mp + c0);
    v4f a1 = *(const v4f*)(mp + c0 + 4);

    float s = (a0.x + a0.y) + (a0.z + a0.w) + (a1.x + a1.y) + (a1.z + a1.w);
    s = wave_sum32(s);
    const float mu = s * (1.0f / C_M);

    float q = (a0.x-mu)*(a0.x-mu) + (a0.y-mu)*(a0.y-mu)
            + (a0.z-mu)*(a0.z-mu) + (a0.w-mu)*(a0.w-mu)
            + (a1.x-mu)*(a1.x-mu) + (a1.y-mu)*(a1.y-mu)
            + (a1.z-mu)*(a1.z-mu) + (a1.w-mu)*(a1.w-mu);
    q = wave_sum32(q);
    const float rs = rsqrtf(q * (1.0f / C_M) + LN_EPS);

    v4f w0 = *(const v4f*)(w + c0),  w1 = *(const v4f*)(w + c0 + 4);
    v4f b0 = *(const v4f*)(b + c0),  b1 = *(const v4f*)(b + c0 + 4);
    v4f m0 = *(const v4f*)(m + (size_t)n * C_M + c0);
    v4f m1 = *(const v4f*)(m + (size_t)n * C_M + c0 + 4);

    v4f o0, o1;
    o0.x = m0.x + ((a0.x - mu) * rs * w0.x + b0.x);
    o0.y = m0.y + ((a0.y - mu) * rs * w0.y + b0.y);
    o0.z = m0.z + ((a0.z - mu) * rs * w0.z + b0.z);
    o0.w = m0.w + ((a0.w - mu) * rs * w0.w + b0.w);
    o1.x = m1.x + ((a1.x - mu) * rs * w1.x + b1.x);
    o1.y = m1.y + ((a1.y - mu) * rs * w1.y + b1.y);
    o1.z = m1.z + ((a1.z - mu) * rs * w1.z + b1.z);
    o1.w = m1.w + ((a1.w - mu) * rs * w1.w + b1.w);

    *(v4f*)(out + (size_t)n * C_M + c0)     = o0;
    *(v4f*)(out + (size_t)n * C_M + c0 + 4) = o1;
}

// ---------------------------------------------------------------------------
// Kernel 3: z_out[i,j,:] = z + emb(bin(i,j)) + LayerNorm(z_prev).
// One block (256 thr = 8 waves) per 16 consecutive pairs (fixed i, 16 j's).
// Phase 1: each wave computes a 16x16 emb tile with V_WMMA_F32_16X16X4_F32
//          (K = 16 = 15 bins + zero pad), C preloaded with lin_b -> exact fp32.
// Phase 2: per-pair layernorm via wave32 shuffle reduction, fused add, NT store.
// ---------------------------------------------------------------------------
__global__ void __launch_bounds__(256)
z_fused_kernel(const float* __restrict__ z, const float* __restrict__ z_prev,
               const float* __restrict__ x_prev,
               const float* __restrict__ ln_w, const float* __restrict__ ln_b,
               const float* __restrict__ lin_w, const float* __restrict__ lin_b,
               float* __restrict__ out) {
    __shared__ __align__(16) float emb_s[16 * C_Z];

    const int tid  = threadIdx.x;
    const int lane = tid & 31;
    const int wv   = tid >> 5;            // wave 0..7
    const int half = lane >> 4;           // 0: lanes 0-15, 1: lanes 16-31
    const int l16  = lane & 15;

    const int i  = blockIdx.x / (NRES / 16);
    const int jt = blockIdx.x % (NRES / 16);

    // ---- bin index for pair (i, jt*16 + l16) ----
    const int j = jt * 16 + l16;
    const float dx = x_prev[3 * i + 0] - x_prev[3 * j + 0];
    const float dy = x_prev[3 * i + 1] - x_prev[3 * j + 1];
    const float dz = x_prev[3 * i + 2] - x_prev[3 * j + 2];
    const float d  = sqrtf(dx * dx + dy * dy + dz * dz);

    int bin = -1;
    {
        const int kk = (int)floorf((d - BIN_START) / BIN_STEP);
        #pragma unroll
        for (int t = -1; t <= 1; ++t) {
            const int k = kk + t;
            if (k >= 0 && k < NBIN) {
                const float lo = BIN_START + BIN_STEP * (float)k;
                const float hi = (k == NBIN - 1) ? 1e8f : (lo + BIN_STEP);
                if (d > lo && d < hi) bin = k;
            }
        }
    }

    // ---- phase 1: emb tile via WMMA f32 16x16x4 ----
    // wave wv owns channels ch = 16*wv + l16; A = onehot (same for all waves)
    const int ch = wv * 16 + l16;
    const float lb = lin_b[ch];
    v8f acc;
    #pragma unroll
    for (int r = 0; r < 8; ++r) acc[r] = lb;

    #pragma unroll
    for (int c4 = 0; c4 < 4; ++c4) {
        v2f A, B;
        #pragma unroll
        for (int v = 0; v < 2; ++v) {
            const int k = c4 * 4 + v + 2 * half;   // ISA f32 A/B 16x4 layout
            A[v] = (bin == k) ? 1.0f : 0.0f;
            B[v] = (k < NBIN) ? lin_w[ch * NBIN + k] : 0.0f;
        }
        acc = __builtin_amdgcn_wmma_f32_16x16x4_f32(
            false, A, false, B, (short)0, acc, false, false);
    }

    // D layout: lane, VGPR r -> pair = r + 8*half, channel = ch
    #pragma unroll
    for (int r = 0; r < 8; ++r)
        emb_s[(r + 8 * half) * C_Z + ch] = acc[r];

    __syncthreads();

    // ---- phase 2: per-pair layernorm + fuse (wave handles 2 pairs) ----
    const int c0 = lane * 4;
    const v4f wz = *(const v4f*)(ln_w + c0);
    const v4f bz = *(const v4f*)(ln_b + c0);

    #pragma unroll
    for (int t = 0; t < 2; ++t) {
        const int p = wv * 2 + t;
        const size_t base = ((size_t)i * NRES + (size_t)(jt * 16 + p)) * C_Z + c0;

        const v4f zp = __builtin_nontemporal_load((const v4f*)(z_prev + base));
        const v4f zz = __builtin_nontemporal_load((const v4f*)(z + base));

        float s = (zp.x + zp.y) + (zp.z + zp.w);
        s = wave_sum32(s);
        const float mu = s * (1.0f / C_Z);

        float q = (zp.x-mu)*(zp.x-mu) + (zp.y-mu)*(zp.y-mu)
                + (zp.z-mu)*(zp.z-mu) + (zp.w-mu)*(zp.w-mu);
        q = wave_sum32(q);
        const float rs = rsqrtf(q * (1.0f / C_Z) + LN_EPS);

        const v4f e = *(const v4f*)(&emb_s[p * C_Z + c0]);

        v4f o;
        o.x = (zz.x + e.x) + ((zp.x - mu) * rs * wz.x + bz.x);
        o.y = (zz.y + e.y) + ((zp.y - mu) * rs * wz.y + bz.y);
        o.z = (zz.z + e.z) + ((zp.z - mu) * rs * wz.z + bz.z);
        o.w = (zz.w + e.w) + ((zp.w - mu) * rs * wz.w + bz.w);

        __builtin_nontemporal_store(o, (v4f*)(out + base));
    }
}

// ---------------------------------------------------------------------------
extern "C" void kernel_launch(void* const* d_in, const int* in_sizes, int n_in,
                              void* d_out, int out_size, void* d_ws, size_t ws_size,
                              hipStream_t stream) {
    (void)in_sizes; (void)n_in; (void)out_size; (void)d_ws; (void)ws_size;

    const float* m      = (const float*)d_in[0];
    const float* z      = (const float*)d_in[1];
    const float* m_prev = (const float*)d_in[2];
    const float* z_prev = (const float*)d_in[3];
    const float* x_prev = (const float*)d_in[4];
    const float* ln_m_w = (const float*)d_in[5];
    const float* ln_m_b = (const float*)d_in[6];
    const float* ln_z_w = (const float*)d_in[7];
    const float* ln_z_b = (const float*)d_in[8];
    const float* lin_w  = (const float*)d_in[9];
    const float* lin_b  = (const float*)d_in[10];

    float* out_m = (float*)d_out;
    float* out_z = out_m + (size_t)SSEQ * NRES * C_M;

    // (SSEQ-1)*NRES*C_M / 4 float4s = 6,242,304 -> / 256 = 24384 blocks (exact)
    m_copy_kernel<<<24384, 256, 0, stream>>>(m, out_m);
    // 768 rows, 8 waves/block -> 96 blocks
    m0_ln_kernel<<<96, 256, 0, stream>>>(m, m_prev, ln_m_w, ln_m_b, out_m);
    // NRES*NRES / 16 pairs-per-block = 36864 blocks
    z_fused_kernel<<<(NRES * NRES) / 16, 256, 0, stream>>>(
        z, z_prev, x_prev, ln_z_w, ln_z_b, lin_w, lin_b, out_z);
}